// ConvolutionalLayer_69286412419079
// MI455X (gfx1250) — compile-verified
//
#include <hip/hip_runtime.h>

typedef __attribute__((ext_vector_type(2))) float v2f;
typedef __attribute__((ext_vector_type(8))) float v8f;

#define NN 512          // nodes per graph
#define DD 64           // feature dim
#define ROWSTR 516      // 512 + 4 pad floats -> conflict-free LDS column reads

// ---------------------------------------------------------------------------
// Kernel A: pre_sup = x @ W   ([32768 x 64] = [32768 x 64] [64 x 64]), fp32 WMMA
// 128 threads = 4 waves; each wave owns one 16-wide column tile of a 16-row slab.
// ---------------------------------------------------------------------------
__global__ __launch_bounds__(128) void presup_gemm_kernel(
    const float* __restrict__ x, const float* __restrict__ W,
    float* __restrict__ psup) {
  const int row0 = blockIdx.x * 16;
  const int wave = threadIdx.x >> 5;
  const int lane = threadIdx.x & 31;
  const int m    = lane & 15;
  const int koff = (lane >> 4) << 1;   // lanes 16-31 hold K+2, K+3
  const int col0 = wave * 16;

  v8f c = {};
  for (int k = 0; k < DD; k += 4) {
    v2f a, b;
    a.x = x[(size_t)(row0 + m) * DD + k + koff];
    a.y = x[(size_t)(row0 + m) * DD + k + koff + 1];
    b.x = W[(size_t)(k + koff) * DD + col0 + m];
    b.y = W[(size_t)(k + koff + 1) * DD + col0 + m];
    c = __builtin_amdgcn_wmma_f32_16x16x4_f32(false, a, false, b,
                                              (short)0, c, false, false);
  }
  // C/D layout: VGPR r -> M=r (lanes 0-15), M=r+8 (lanes 16-31); N = lane%16
  const int mbase = (lane < 16) ? 0 : 8;
  for (int r = 0; r < 8; ++r)
    psup[(size_t)(row0 + mbase + r) * DD + col0 + m] = c[r];
}

// ---------------------------------------------------------------------------
// Kernel B: fused  scores -> softmax -> 0.5*(A+I) -> (adj_norm @ pre_sup) -> relu
// One block per (batch b, 16-row tile). 256 threads = 8 waves.
//   Phase 1: stream edge_features slice (16*512*8 f32 = 256 KB), dot with coef.
//   Phase 2: per-row softmax with wave32 shuffle reductions (values in regs).
//            Degree of A+I is exactly 2 (softmax rows sum to 1), so
//            D^-1/2 (A+I) D^-1/2 == 0.5*(softmax + I).
//   Phase 3: 16x64 output tile via V_WMMA_F32_16X16X4_F32, K=512 split over
//            two wave groups, partials reduced through LDS, ReLU on store.
// ---------------------------------------------------------------------------
__global__ __launch_bounds__(256) void fused_gcn_kernel(
    const float* __restrict__ ef, const float* __restrict__ psup,
    const float* __restrict__ coef, float* __restrict__ out) {
  __shared__ float sc[16 * ROWSTR];   // scores, then adj_norm (33 KB)
  __shared__ float cbuf[4 * 256];     // cross-wave-group C partials (4 KB)

  const int b    = blockIdx.x >> 5;          // 32 row-tiles per batch
  const int i0   = (blockIdx.x & 31) << 4;
  const int tid  = threadIdx.x;
  const int wave = tid >> 5;
  const int lane = tid & 31;

  const float c0 = coef[0], c1 = coef[1], c2 = coef[2], c3 = coef[3];
  const float c4 = coef[4], c5 = coef[5], c6 = coef[6], c7 = coef[7];

  // ---- Phase 1: scores[i][j] = dot(ef[b,i0+i,j,:], coef) ----
  const float* efb = ef + (size_t)(b * NN + i0) * NN * 8;
  for (int s = tid; s < 16 * NN; s += 256) {
    const int i = s >> 9;
    const int j = s & (NN - 1);
    const float* p = efb + ((size_t)i * NN + j) * 8;
    const float4 u = *(const float4*)p;
    const float4 v = *(const float4*)(p + 4);
    sc[i * ROWSTR + j] = u.x * c0 + u.y * c1 + u.z * c2 + u.w * c3 +
                         v.x * c4 + v.y * c5 + v.z * c6 + v.w * c7;
  }
  __syncthreads();

  // ---- Phase 2: softmax per row; write adj_norm = 0.5*(softmax + I) ----
  for (int rr = 0; rr < 2; ++rr) {
    const int r = wave * 2 + rr;           // 8 waves x 2 rows = 16 rows
    float v[16];
    float mx = -3.402823466e38f;
    for (int t = 0; t < 16; ++t) {
      v[t] = sc[r * ROWSTR + lane + 32 * t];
      mx = fmaxf(mx, v[t]);
    }
    for (int off = 16; off; off >>= 1) mx = fmaxf(mx, __shfl_xor(mx, off, 32));
    float sm = 0.f;
    for (int t = 0; t < 16; ++t) {
      v[t] = __expf(v[t] - mx);
      sm += v[t];
    }
    for (int off = 16; off; off >>= 1) sm += __shfl_xor(sm, off, 32);
    const float inv = 0.5f / sm;           // includes both D^-1/2 factors
    const int gi = i0 + r;
    for (int t = 0; t < 16; ++t) {
      const int j = lane + 32 * t;
      sc[r * ROWSTR + j] = v[t] * inv + ((j == gi) ? 0.5f : 0.f);
    }
  }
  __syncthreads();

  // ---- Phase 3: out[16,64] = adjn[16,512] @ psup_b[512,64] ----
  const float* pb   = psup + (size_t)(b * NN) * DD;  // L2-hot across 32 blocks
  const int dtile   = (wave & 3) * 16;
  const int kbase   = (wave >> 2) * 256;             // K split across wave groups
  const int m       = lane & 15;
  const int koff    = (lane >> 4) << 1;

  v8f acc = {};
  for (int k = kbase; k < kbase + 256; k += 4) {
    v2f a, bv;
    a.x  = sc[m * ROWSTR + k + koff];
    a.y  = sc[m * ROWSTR + k + koff + 1];
    bv.x = pb[(size_t)(k + koff) * DD + dtile + m];
    bv.y = pb[(size_t)(k + koff + 1) * DD + dtile + m];
    acc = __builtin_amdgcn_wmma_f32_16x16x4_f32(false, a, false, bv,
                                                (short)0, acc, false, false);
  }

  if (wave >= 4) {
    for (int r = 0; r < 8; ++r)
      cbuf[(wave - 4) * 256 + r * 32 + lane] = acc[r];
  }
  __syncthreads();
  if (wave < 4) {
    const int mbase = i0 + ((lane < 16) ? 0 : 8);
    for (int r = 0; r < 8; ++r) {
      float o = acc[r] + cbuf[wave * 256 + r * 32 + lane];
      o = fmaxf(o, 0.f);
      out[(size_t)(b * NN + mbase + r) * DD + dtile + m] = o;
    }
  }
}

extern "C" void kernel_launch(void* const* d_in, const int* in_sizes, int n_in,
                              void* d_out, int out_size, void* d_ws, size_t ws_size,
                              hipStream_t stream) {
  (void)in_sizes; (void)n_in; (void)out_size; (void)ws_size;
  const float* ef   = (const float*)d_in[0];   // [64,512,512,8]
  const float* x    = (const float*)d_in[1];   // [32768,64]
  const float* W    = (const float*)d_in[2];   // [64,64]
  const float* coef = (const float*)d_in[3];   // [8,1]
  float* out  = (float*)d_out;                 // [32768,64]
  float* psup = (float*)d_ws;                  // 8 MB scratch: pre_sup

  presup_gemm_kernel<<<32768 / 16, 128, 0, stream>>>(x, W, psup);
  fused_gcn_kernel<<<64 * 32, 256, 0, stream>>>(ef, psup, coef, out);
}